// Loss_OMAD_PriorNet_3204045603277
// MI455X (gfx1250) — compile-verified
//
#include <hip/hip_runtime.h>
#include <math.h>

typedef float v2f __attribute__((ext_vector_type(2)));
typedef float v8f __attribute__((ext_vector_type(8)));

#define BSZ   32
#define KSZ   2
#define PSZ   256
#define NSZ   8192
#define COEFN (32 * 512 * 10)

// d_ws layout (floats)
#define SURF_OFF  0      // 1024 partials (B * 32 strips)
#define CHF_OFF   1024   // 64 partials (B*K)
#define SEP_OFF   1088   // 64 partials
#define COV_OFF   1152   // 64 partials
#define JOINT_OFF 1216   // 2 values: loc_sum, cos_sum
#define REG_OFF   1218   // 1 value

__device__ __forceinline__ float smooth_l1(float pred, float target) {
    float d = fabsf(pred - target);
    return (d < 0.1f) ? (0.5f * d * d / 0.1f) : (d - 0.05f);
}

// blockDim.x == 256 tree reduction, fixed order -> deterministic.
// OP: 0=max, 1=min, 2=sum
template <int OP>
__device__ float bred(float v, float* red) {
    const int t = threadIdx.x;
    red[t] = v;
    __syncthreads();
    for (int s = 128; s > 0; s >>= 1) {
        if (t < s) {
            float o = red[t + s];
            if (OP == 0)      red[t] = fmaxf(red[t], o);
            else if (OP == 1) red[t] = fminf(red[t], o);
            else              red[t] = red[t] + o;
        }
        __syncthreads();
    }
    float r = red[0];
    __syncthreads();
    return r;
}

// ---------------------------------------------------------------------------
// Surface loss: min over cloud of pdist(nodes, cloud), WMMA f32 16x16x4.
// One wave = (batch b, strip s of 16 nodes). 4 waves / block, 256 blocks.
// ---------------------------------------------------------------------------
__global__ __launch_bounds__(128) void surf_kernel(
    const float* __restrict__ nodes,  // (B, 512, 3)
    const float* __restrict__ cloud,  // (B, 8192, 3)
    float* __restrict__ ws_surf) {
    const int lane = threadIdx.x & 31;
    const int widx = threadIdx.x >> 5;
    const int w    = blockIdx.x * 4 + widx;   // 0..1023
    const int b    = w >> 5;
    const int s    = w & 31;
    const int hi   = lane >> 4;               // half-wave select
    const int l15  = lane & 15;

    // A matrix: 16 nodes x 4 coords (K=3 padded with 0).
    // ISA layout: VGPR0 lanes0-15 = K0, lanes16-31 = K2; VGPR1 = K1 / K3.
    const float* np = nodes + ((size_t)b * 512 + (size_t)s * 16 + l15) * 3;
    float nx = np[0], ny = np[1], nz = np[2];
    v2f A;
    A.x = hi ? nz : nx;
    A.y = hi ? 0.0f : ny;
    float an = nx * nx + ny * ny + nz * nz;   // ||node[l15]||^2

    // anr[r] = ||node[r + 8*hi]||^2 (constant across cloud tiles)
    float anr[8];
#pragma unroll
    for (int r = 0; r < 8; ++r)
        anr[r] = __shfl(an, r + 8 * hi, 32);

    v8f md;
#pragma unroll
    for (int r = 0; r < 8; ++r) md[r] = 3.4e38f;

    const float* cb = cloud + (size_t)b * NSZ * 3;
#pragma unroll 4
    for (int n0 = 0; n0 < NSZ; n0 += 16) {
        // B matrix: 4 x 16 cloud coords; each lane loads its column's point.
        const float* cp = cb + (size_t)(n0 + l15) * 3;
        float cx = cp[0], cy = cp[1], cz = cp[2];
        v2f Bm;
        Bm.x = hi ? cz : cx;
        Bm.y = hi ? 0.0f : cy;
        float bn = cx * cx + cy * cy + cz * cz;

        v8f c = {};
        c = __builtin_amdgcn_wmma_f32_16x16x4_f32(
            false, A, false, Bm, (short)0, c, false, false);

#pragma unroll
        for (int r = 0; r < 8; ++r) {
            float d2 = fmaf(-2.0f, c[r], anr[r] + bn);  // |a|^2+|b|^2-2ab
            d2 = fmaxf(d2, 1e-12f);
            md[r] = fminf(md[r], d2);
        }
    }

    // min across the 16 lanes of each half, then sum sqrt over the 16 rows
    float sum = 0.0f;
#pragma unroll
    for (int r = 0; r < 8; ++r) {
        float v = md[r];
        v = fminf(v, __shfl_xor(v, 1, 32));
        v = fminf(v, __shfl_xor(v, 2, 32));
        v = fminf(v, __shfl_xor(v, 4, 32));
        v = fminf(v, __shfl_xor(v, 8, 32));
        sum += sqrtf(v);
    }
    sum += __shfl_xor(sum, 16, 32);           // combine M=0..7 and M=8..15
    if (lane == 0) ws_surf[w] = sum;
}

// ---------------------------------------------------------------------------
// Chamfer (fwd+bwd) + separation. One block per (b,k).
// ---------------------------------------------------------------------------
__global__ __launch_bounds__(256) void chamfer_sep_kernel(
    const float* __restrict__ kps,    // (B,K,P,3)
    const float* __restrict__ nodes,  // (B,K,P,3)
    const float* __restrict__ scale,  // (B,K,3)
    float* __restrict__ ws_chf, float* __restrict__ ws_sep) {
    __shared__ float skp[PSZ][3];
    __shared__ float snd[PSZ][3];
    __shared__ float red[256];
    const int t  = threadIdx.x;
    const int bk = blockIdx.x;

    const float* kb = kps   + (size_t)bk * PSZ * 3;
    const float* nb = nodes + (size_t)bk * PSZ * 3;
    skp[t][0] = kb[t * 3 + 0]; skp[t][1] = kb[t * 3 + 1]; skp[t][2] = kb[t * 3 + 2];
    snd[t][0] = nb[t * 3 + 0]; snd[t][1] = nb[t * 3 + 1]; snd[t][2] = nb[t * 3 + 2];
    __syncthreads();

    float sx = scale[bk * 3 + 0], sy = scale[bk * 3 + 1], sz = scale[bk * 3 + 2];
    float thr = sqrtf(sx * sx + sy * sy + sz * sz) * 0.125f;

    float kx = skp[t][0], ky = skp[t][1], kz = skp[t][2];
    float nx = snd[t][0], ny = snd[t][1], nz = snd[t][2];

    float minf = 3.4e38f, minb = 3.4e38f, sep = 0.0f;
    for (int j = 0; j < PSZ; ++j) {
        float jx = snd[j][0], jy = snd[j][1], jz = snd[j][2];
        float px = skp[j][0], py = skp[j][1], pz = skp[j][2];
        // fwd: kp[t] vs node[j]
        float dx = kx - jx, dy = ky - jy, dz = kz - jz;
        minf = fminf(minf, dx * dx + dy * dy + dz * dz);
        // bwd: node[t] vs kp[j]
        float ex = nx - px, ey = ny - py, ez = nz - pz;
        minb = fminf(minb, ex * ex + ey * ey + ez * ez);
        // sep: node[t] vs node[j], j != t
        float fx = nx - jx, fy = ny - jy, fz = nz - jz;
        float f2 = fx * fx + fy * fy + fz * fz;
        if (j != t) sep += fmaxf(0.0f, thr - sqrtf(f2));
    }
    float chfp = sqrtf(fmaxf(minf, 1e-12f)) + sqrtf(fmaxf(minb, 1e-12f));

    float chf_sum = bred<2>(chfp, red);
    float sep_sum = bred<2>(sep, red);
    if (t == 0) { ws_chf[bk] = chf_sum; ws_sep[bk] = sep_sum; }
}

// ---------------------------------------------------------------------------
// Coverage loss. One block per (b,k).
// ---------------------------------------------------------------------------
__global__ __launch_bounds__(256) void cov_kernel(
    const float* __restrict__ cloud,  // (B,N,3)
    const int* __restrict__ cls,      // (B,N)
    const float* __restrict__ nodes,  // (B,K,P,3)
    float* __restrict__ ws_cov) {
    __shared__ float red[256];
    const int bk = blockIdx.x;
    const int b = bk >> 1, k = bk & 1;
    const int t = threadIdx.x;

    float pmax0 = -1e9f, pmax1 = -1e9f, pmax2 = -1e9f;
    float pmin0 =  1e9f, pmin1 =  1e9f, pmin2 =  1e9f;
    const float* cb = cloud + (size_t)b * NSZ * 3;
    const int*   lb = cls   + (size_t)b * NSZ;
    for (int n = t; n < NSZ; n += 256) {
        if (lb[n] == k) {
            float x = cb[n * 3 + 0], y = cb[n * 3 + 1], z = cb[n * 3 + 2];
            pmax0 = fmaxf(pmax0, x); pmax1 = fmaxf(pmax1, y); pmax2 = fmaxf(pmax2, z);
            pmin0 = fminf(pmin0, x); pmin1 = fminf(pmin1, y); pmin2 = fminf(pmin2, z);
        }
    }
    const float* nb = nodes + (size_t)bk * PSZ * 3;
    float kx = nb[t * 3 + 0], ky = nb[t * 3 + 1], kz = nb[t * 3 + 2];

    float PM0 = bred<0>(pmax0, red), PM1 = bred<0>(pmax1, red), PM2 = bred<0>(pmax2, red);
    float Pm0 = bred<1>(pmin0, red), Pm1 = bred<1>(pmin1, red), Pm2 = bred<1>(pmin2, red);
    float KM0 = bred<0>(kx, red),    KM1 = bred<0>(ky, red),    KM2 = bred<0>(kz, red);
    float Km0 = bred<1>(kx, red),    Km1 = bred<1>(ky, red),    Km2 = bred<1>(kz, red);

    if (t == 0) {
        float covp = 0.0f;
        covp += 0.5f * (smooth_l1(KM0, PM0) + smooth_l1(Km0, Pm0));
        covp += 0.5f * (smooth_l1(KM1, PM1) + smooth_l1(Km1, Pm1));
        covp += 0.5f * (smooth_l1(KM2, PM2) + smooth_l1(Km2, Pm2));
        ws_cov[bk] = covp;
    }
}

// ---------------------------------------------------------------------------
// Joint loss (block 0) + coefficient regularizer (block 1).
// ---------------------------------------------------------------------------
__global__ __launch_bounds__(256) void joint_reg_kernel(
    const float* __restrict__ pjl, const float* __restrict__ pja,
    const float* __restrict__ gjl, const float* __restrict__ gja,
    const float* __restrict__ coefs,
    float* __restrict__ ws_joint, float* __restrict__ ws_reg) {
    __shared__ float red[256];
    const int t = threadIdx.x;
    if (blockIdx.x == 0) {
        float loc = 0.0f, cosv = 0.0f;
        if (t < BSZ) {  // K-1 == 1 joint per batch
            float ax = pja[t * 3 + 0], ay = pja[t * 3 + 1], az = pja[t * 3 + 2];
            float gx = gja[t * 3 + 0], gy = gja[t * 3 + 1], gz = gja[t * 3 + 2];
            float dot = ax * gx + ay * gy + az * gz;
            float na = sqrtf(ax * ax + ay * ay + az * az);
            float ng = sqrtf(gx * gx + gy * gy + gz * gz);
            cosv = dot / fmaxf(na * ng, 1e-8f);

            float ix = gx / ng, iy = gy / ng, iz = gz / ng;      // ngt
            float px = gjl[t * 3 + 0], py = gjl[t * 3 + 1], pz = gjl[t * 3 + 2];
            float qx = px + ix, qy = py + iy, qz = pz + iz;
            float rx = pjl[t * 3 + 0], ry = pjl[t * 3 + 1], rz = pjl[t * 3 + 2];
            float xx = px - qx, xy = py - qy, xz = pz - qz;       // p - q
            float num = (rx - qx) * xx + (ry - qy) * xy + (rz - qz) * xz;
            float den = xx * xx + xy * xy + xz * xz;
            float tt  = num / den;
            float vx = tt * xx + (qx - rx);
            float vy = tt * xy + (qy - ry);
            float vz = tt * xz + (qz - rz);
            loc = sqrtf(vx * vx + vy * vy + vz * vz);
        }
        float loc_sum = bred<2>(loc, red);
        float cos_sum = bred<2>(cosv, red);
        if (t == 0) { ws_joint[0] = loc_sum; ws_joint[1] = cos_sum; }
    } else {
        float s = 0.0f;
        for (int i = t; i < COEFN; i += 256) { float c = coefs[i]; s += c * c; }
        float reg_sum = bred<2>(s, red);
        if (t == 0) ws_reg[0] = reg_sum;
    }
}

// ---------------------------------------------------------------------------
// Finalize: deterministic global reduction + weighted combine.
// ---------------------------------------------------------------------------
__global__ __launch_bounds__(256) void finalize_kernel(
    const float* __restrict__ ws, float* __restrict__ out) {
    __shared__ float red[256];
    const int t = threadIdx.x;
    float v = 0.0f;
    for (int i = t; i < 1024; i += 256) v += ws[SURF_OFF + i];
    float surf_sum = bred<2>(v, red);
    float chf_sum  = bred<2>((t < 64) ? ws[CHF_OFF + t] : 0.0f, red);
    float sep_sum  = bred<2>((t < 64) ? ws[SEP_OFF + t] : 0.0f, red);
    float cov_sum  = bred<2>((t < 64) ? ws[COV_OFF + t] : 0.0f, red);

    if (t == 0) {
        float loc_sum = ws[JOINT_OFF + 0];
        float cos_sum = ws[JOINT_OFF + 1];
        float reg_sum = ws[REG_OFF];

        float chf   = chf_sum  / 16384.0f;                 // B*K*P
        float cov   = cov_sum  / 192.0f;                   // B*K*3
        float surf  = surf_sum / 16384.0f;                 // B*(K*P)
        float sep   = sep_sum  / 4177920.0f;               // B*K*P*(P-1)
        float joint = loc_sum / 32.0f + 0.5f * (1.0f - cos_sum / 32.0f);
        float reg   = reg_sum / (float)COEFN;

        out[0] = chf * 1.0f + cov * 1.0f + surf * 5.0f +
                 joint * 1.0f + reg * 0.01f + sep * 2.0f;
    }
}

extern "C" void kernel_launch(void* const* d_in, const int* in_sizes, int n_in,
                              void* d_out, int out_size, void* d_ws, size_t ws_size,
                              hipStream_t stream) {
    (void)in_sizes; (void)n_in; (void)out_size; (void)ws_size;
    const float* coefs = (const float*)d_in[0];
    const float* kps   = (const float*)d_in[1];
    const float* nodes = (const float*)d_in[2];
    const float* cloud = (const float*)d_in[3];
    const int*   ccls  = (const int*)d_in[4];
    const float* pjl   = (const float*)d_in[5];
    const float* pja   = (const float*)d_in[6];
    const float* gjl   = (const float*)d_in[7];
    const float* gja   = (const float*)d_in[8];
    const float* gps   = (const float*)d_in[9];
    float* out = (float*)d_out;
    float* ws  = (float*)d_ws;

    surf_kernel<<<256, 128, 0, stream>>>(nodes, cloud, ws + SURF_OFF);
    chamfer_sep_kernel<<<64, 256, 0, stream>>>(kps, nodes, gps,
                                               ws + CHF_OFF, ws + SEP_OFF);
    cov_kernel<<<64, 256, 0, stream>>>(cloud, ccls, nodes, ws + COV_OFF);
    joint_reg_kernel<<<2, 256, 0, stream>>>(pjl, pja, gjl, gja, coefs,
                                            ws + JOINT_OFF, ws + REG_OFF);
    finalize_kernel<<<1, 256, 0, stream>>>(ws, out);
}